// SparseConvolutionBase_19258633356183
// MI455X (gfx1250) — compile-verified
//
#include <hip/hip_runtime.h>

typedef __attribute__((ext_vector_type(2))) float v2f;
typedef __attribute__((ext_vector_type(8))) float v8f;

#define N_IN   100000
#define N_OUT  100000
#define KVOL   27
#define MPAIR  30000
#define CIN    256
#define COUT   256

#define TILE_M 32
#define KCHUNK 32
#define SA     (KCHUNK + 2)   // padded LDS row stride (floats) for A tile
#define SB     (COUT + 2)     // padded LDS row stride (floats) for B chunk

// ---------------------------------------------------------------------------
// out[i, :] = bias[:]  for all N_OUT rows (reference adds bias everywhere)
// ---------------------------------------------------------------------------
__global__ __launch_bounds__(256) void bias_init_kernel(
    const float* __restrict__ bias, float* __restrict__ out) {
  int i = blockIdx.x * blockDim.x + threadIdx.x;  // float4 index
  const int total4 = N_OUT * COUT / 4;
  if (i >= total4) return;
  int c4 = i & (COUT / 4 - 1);
  ((float4*)out)[i] = ((const float4*)bias)[c4];
}

// Guaranteed single-instruction f32 atomic add (no CAS loop).
__device__ inline void global_atomic_fadd(float* p, float v) {
  asm volatile("global_atomic_add_f32 %0, %1, off" :: "v"(p), "v"(v) : "memory");
}

// ---------------------------------------------------------------------------
// One block = (kernel offset k, 32-row tile of M). 8 waves; wave w computes a
// 16(M) x 64(N) slab: m-subtile = w&1, n-base = (w>>1)*64, 4 WMMA n-tiles.
// ---------------------------------------------------------------------------
__global__ __launch_bounds__(256) void sparse_conv_wmma_kernel(
    const float* __restrict__ features,
    const float* __restrict__ kern,
    const int*   __restrict__ in_map,
    const int*   __restrict__ out_map,
    float*       __restrict__ out) {
  __shared__ float A_lds[TILE_M * SA];   // gathered features chunk: 32 x 32
  __shared__ float B_lds[KCHUNK * SB];   // kernel chunk: 32 x 256
  __shared__ int   in_rows[TILE_M];
  __shared__ int   out_rows[TILE_M];

  const int tilesM = (MPAIR + TILE_M - 1) / TILE_M;   // 938
  const int k   = blockIdx.x / tilesM;
  const int mt  = blockIdx.x % tilesM;
  const int m0  = mt * TILE_M;
  const int tid = threadIdx.x;

  // Stage gather/scatter row indices (sentinel -1 for the partial last tile).
  if (tid < TILE_M) {
    int mi = m0 + tid;
    bool ok = mi < MPAIR;
    in_rows[tid]  = ok ? in_map[k * MPAIR + mi]  : -1;
    out_rows[tid] = ok ? out_map[k * MPAIR + mi] : -1;
  }

  const int wave = tid >> 5;
  const int lane = tid & 31;
  const int msub = wave & 1;           // rows msub*16 .. msub*16+15
  const int n0   = (wave >> 1) * 64;   // 64-column slab, 4 n-tiles of 16

  v8f acc0 = {}, acc1 = {}, acc2 = {}, acc3 = {};

  const float* kbase = kern + (size_t)k * CIN * COUT;

  // A-fragment addressing (f32 16x16x4 layout): lane<16 -> K, K+1 ; lane>=16 -> K+2, K+3
  const int aRow = msub * 16 + (lane & 15);
  const int kHi  = (lane >> 4) * 2;
  const int bCol = lane & 15;

  for (int kk = 0; kk < CIN; kk += KCHUNK) {
    __syncthreads();  // previous chunk's compute done; index stage visible

    // Gather A chunk: 32 rows x 32 cols = 256 float4 (1 per thread)
    {
      int row = tid >> 3;                // KCHUNK/4 = 8 float4 per row
      int c4  = tid & 7;
      int src = in_rows[row];
      float4 val = make_float4(0.f, 0.f, 0.f, 0.f);
      if (src >= 0)
        val = ((const float4*)(features + (size_t)src * CIN + kk))[c4];
      *(float4*)(&A_lds[row * SA + c4 * 4]) = val;
    }
    // Load B chunk: 32 rows x 256 cols = 2048 float4 (8 per thread)
    {
      #pragma unroll
      for (int it = 0; it < 8; ++it) {
        int f   = tid + it * 256;
        int row = f >> 6;                // COUT/4 = 64 float4 per row
        int c4  = f & 63;
        float4 val = ((const float4*)(kbase + (size_t)(kk + row) * COUT))[c4];
        *(float4*)(&B_lds[row * SB + c4 * 4]) = val;
      }
    }
    __syncthreads();

    #pragma unroll
    for (int kkk = 0; kkk < KCHUNK; kkk += 4) {
      // A fragment: 16x4 f32, ds_load_b64 of {K, K+1} at lane-dependent row
      v2f a = *(const v2f*)(&A_lds[aRow * SA + kkk + kHi]);
      const float* bp = &B_lds[(kkk + kHi) * SB + bCol];

      v2f b0 = { bp[n0 +  0], bp[SB + n0 +  0] };
      acc0 = __builtin_amdgcn_wmma_f32_16x16x4_f32(false, a, false, b0,
                                                   (short)0, acc0, false, false);
      v2f b1 = { bp[n0 + 16], bp[SB + n0 + 16] };
      acc1 = __builtin_amdgcn_wmma_f32_16x16x4_f32(false, a, false, b1,
                                                   (short)0, acc1, false, false);
      v2f b2 = { bp[n0 + 32], bp[SB + n0 + 32] };
      acc2 = __builtin_amdgcn_wmma_f32_16x16x4_f32(false, a, false, b2,
                                                   (short)0, acc2, false, false);
      v2f b3 = { bp[n0 + 48], bp[SB + n0 + 48] };
      acc3 = __builtin_amdgcn_wmma_f32_16x16x4_f32(false, a, false, b3,
                                                   (short)0, acc3, false, false);
    }
  }

  // Scatter-add: C/D layout -> lane covers col (lane&15), rows r + 8*(lane>>4)
  const int rHi     = (lane >> 4) * 8;
  const int colBase = n0 + (lane & 15);
  #pragma unroll
  for (int r = 0; r < 8; ++r) {
    int m    = msub * 16 + r + rHi;
    int orow = out_rows[m];
    if (orow >= 0) {
      float* p = out + (size_t)orow * COUT + colBase;
      global_atomic_fadd(p +  0, acc0[r]);
      global_atomic_fadd(p + 16, acc1[r]);
      global_atomic_fadd(p + 32, acc2[r]);
      global_atomic_fadd(p + 48, acc3[r]);
    }
  }
}

extern "C" void kernel_launch(void* const* d_in, const int* in_sizes, int n_in,
                              void* d_out, int out_size, void* d_ws, size_t ws_size,
                              hipStream_t stream) {
  const float* features = (const float*)d_in[0];  // [N_IN, C_IN]
  const float* kern     = (const float*)d_in[1];  // [K, C_IN, C_OUT]
  const float* bias     = (const float*)d_in[2];  // [1, C_OUT]
  const int*   in_map   = (const int*)d_in[3];    // [K, M]
  const int*   out_map  = (const int*)d_in[4];    // [K, M]
  float* out = (float*)d_out;                     // [N_OUT, C_OUT]

  const int total4 = N_OUT * COUT / 4;
  bias_init_kernel<<<(total4 + 255) / 256, 256, 0, stream>>>(bias, out);

  const int tilesM = (MPAIR + TILE_M - 1) / TILE_M;  // 938
  sparse_conv_wmma_kernel<<<KVOL * tilesM, 256, 0, stream>>>(
      features, kern, in_map, out_map, out);
}